// Screen_39608188403925
// MI455X (gfx1250) — compile-verified
//
#include <hip/hip_runtime.h>
#include <hip/hip_bf16.h>
#include <stdint.h>

#define TPB 256

namespace cfg {
constexpr int      NBW   = 120, NBH = 68, NB = 8160;
constexpr int      NPTS  = 500000;
constexpr int      KWIN  = 5;
constexpr unsigned EI    = 12500000u;                       // NPTS * KWIN * KWIN
constexpr int      NBLK_A = (NPTS + TPB - 1) / TPB;         // 1954
constexpr int      IPT   = 16;
constexpr int      CHUNK = TPB * IPT;                       // 4096
constexpr int      NBLK_R = (int)((EI + CHUNK - 1) / CHUNK);// 3052
constexpr unsigned HIST_LEN = 256u * (unsigned)NBLK_R;      // 781312
constexpr int      PASSES = 6;
constexpr int      SHIFT0 = 16;                             // sort key bits 16..63
}

typedef unsigned int v4u __attribute__((ext_vector_type(4)));
typedef int          v8i __attribute__((ext_vector_type(8)));
typedef int          v4i __attribute__((ext_vector_type(4)));

__device__ __forceinline__ unsigned ballot32(bool b) {
  return (unsigned)__builtin_amdgcn_ballot_w32(b);
}

// Tensor Data Mover: contiguous 1-row tile of `tile8` 8-byte elements,
// global -> LDS.  `tens8` = valid element count (OOB reads return zero),
// implementing the tail guard in hardware.  D# packing per ISA 8.3/8.4.
__device__ __forceinline__ void tdm_load_1d_b64(void* lds_dst, const void* gsrc,
                                                unsigned tile8, unsigned tens8) {
  const unsigned long long ga = (unsigned long long)(uintptr_t)gsrc;
  const unsigned lds = (unsigned)(uintptr_t)lds_dst;
  v4u g0;
  g0.x = 1u;                                                // count=1, user mode
  g0.y = lds;                                               // lds_addr
  g0.z = (unsigned)(ga & 0xFFFFFFFFull);                    // global_addr[31:0]
  g0.w = (unsigned)((ga >> 32) & 0x01FFFFFFull) | (2u << 30); // ga[56:32], type=2
  v8i g1;
  g1[0] = (int)(3u << 16);                                  // data_size=3 (8B)
  g1[1] = (int)((tens8 & 0xFFFFu) << 16);                   // tensor_dim0[15:0]
  g1[2] = (int)(((tens8 >> 16) & 0xFFFFu) | (1u << 16));    // tensor_dim0[31:16], tensor_dim1=1
  g1[3] = (int)((tile8 & 0xFFFFu) << 16);                   // tile_dim0
  g1[4] = 1;                                                // tile_dim1=1, tile_dim2=0
  g1[5] = (int)tile8;                                       // tensor_dim0_stride[31:0]
  g1[6] = 0;
  g1[7] = 0;
  const v4i z4 = {0, 0, 0, 0};
#if __clang_major__ >= 23
  const v8i z8 = {0, 0, 0, 0, 0, 0, 0, 0};
  __builtin_amdgcn_tensor_load_to_lds(g0, g1, z4, z4, z8, 0);
#else
  __builtin_amdgcn_tensor_load_to_lds(g0, g1, z4, z4, 0);
#endif
}

struct Rect { int xmn, ymn, xmx, ymx; };

__device__ __forceinline__ Rect tile_rect(float x, float y, float r) {
  Rect q;
  q.xmn = min(max((int)floorf((x - r) * 0.0625f), 0), cfg::NBW);
  q.ymn = min(max((int)floorf((y - r) * 0.0625f), 0), cfg::NBH);
  q.xmx = min(max((int)floorf((x + r) * 0.0625f) + 1, 0), cfg::NBW);
  q.ymx = min(max((int)floorf((y + r) * 0.0625f) + 1, 0), cfg::NBH);
  return q;
}

// ---- K0: zero the global tile-count histogram ------------------------------
__global__ void __launch_bounds__(TPB)
k_zero(unsigned* __restrict__ g_counts) {
  const int i = blockIdx.x * TPB + threadIdx.x;
  if (i < cfg::NB) g_counts[i] = 0u;
}

// ---- K1: per-tile histogram (full 8160-bin LDS hist) + per-block valid count
__global__ void __launch_bounds__(TPB)
k_tile_hist(const float* __restrict__ pos2d, const float* __restrict__ radius,
            unsigned* __restrict__ g_counts, unsigned* __restrict__ g_blk) {
  __shared__ unsigned sh[cfg::NB];     // 32.6 KB of the 320 KB LDS
  __shared__ unsigned red[TPB];
  for (int j = threadIdx.x; j < cfg::NB; j += TPB) sh[j] = 0u;
  __syncthreads();
  const int p = blockIdx.x * TPB + threadIdx.x;
  unsigned nv = 0;
  if (p < cfg::NPTS) {
    __builtin_prefetch(&pos2d[2 * p], 0, 0);   // -> global_prefetch_b8
    const float x = pos2d[2 * p], y = pos2d[2 * p + 1], r = radius[p];
    const Rect q = tile_rect(x, y, r);
#pragma unroll
    for (int dx = 0; dx < cfg::KWIN; ++dx) {
      const int xi = q.xmn + dx;
      const bool vx = xi < q.xmx;
#pragma unroll
      for (int dy = 0; dy < cfg::KWIN; ++dy) {
        const int yi = q.ymn + dy;
        if (vx && (yi < q.ymx)) { atomicAdd(&sh[xi * cfg::NBH + yi], 1u); ++nv; }
      }
    }
  }
  __syncthreads();
  for (int j = threadIdx.x; j < cfg::NB; j += TPB) {
    const unsigned c = sh[j];
    if (c) atomicAdd(&g_counts[j], c);
  }
  red[threadIdx.x] = nv;
  __syncthreads();
  for (int off = TPB / 2; off > 0; off >>= 1) {
    if (threadIdx.x < off) red[threadIdx.x] += red[threadIdx.x + off];
    __syncthreads();
  }
  if (threadIdx.x == 0) g_blk[blockIdx.x] = red[0];
}

// ---- K2: exclusive scan of per-block valid counts (in place) + total M -----
__global__ void __launch_bounds__(TPB)
k_scan_blocks(unsigned* __restrict__ g_blk, unsigned* __restrict__ g_M) {
  __shared__ unsigned s[TPB];
  const int t = threadIdx.x;
  unsigned carry = 0;
  for (int base = 0; base < cfg::NBLK_A; base += TPB) {
    const int i = base + t;
    const unsigned v = (i < cfg::NBLK_A) ? g_blk[i] : 0u;
    s[t] = v;
    __syncthreads();
    for (int off = 1; off < TPB; off <<= 1) {
      const unsigned u = (t >= off) ? s[t - off] : 0u;
      __syncthreads();
      s[t] += u;
      __syncthreads();
    }
    if (i < cfg::NBLK_A) g_blk[i] = carry + s[t] - v;
    const unsigned tot = s[TPB - 1];
    __syncthreads();
    carry += tot;
  }
  if (t == 0) *g_M = carry;
}

// ---- K3: exclusive scan of tile counts -> tile_count output (int32) --------
__global__ void __launch_bounds__(TPB)
k_scan_tiles(const unsigned* __restrict__ g_counts, int* __restrict__ tile_count) {
  __shared__ unsigned s[TPB];
  const int t = threadIdx.x;
  unsigned carry = 0;
  for (int base = 0; base < cfg::NB; base += TPB) {
    const int i = base + t;
    const unsigned v = (i < cfg::NB) ? g_counts[i] : 0u;
    s[t] = v;
    __syncthreads();
    for (int off = 1; off < TPB; off <<= 1) {
      const unsigned u = (t >= off) ? s[t - off] : 0u;
      __syncthreads();
      s[t] += u;
      __syncthreads();
    }
    if (i < cfg::NB) tile_count[i] = (int)(carry + s[t] - v);
    const unsigned tot = s[TPB - 1];
    __syncthreads();
    carry += tot;
  }
}

// ---- K4: stable compaction -> packed 64-bit keys ---------------------------
// key = tid(13) << 51 | ~mono(depth)(32) << 19 | pid(19)
__global__ void __launch_bounds__(TPB)
k_emit(const float* __restrict__ pos2d, const float* __restrict__ radius,
       const float* __restrict__ depth, const unsigned* __restrict__ g_blkoff,
       unsigned long long* __restrict__ keys) {
  __shared__ unsigned s[TPB];
  const int t = threadIdx.x;
  const int p = blockIdx.x * TPB + t;
  Rect q; q.xmn = q.ymn = q.xmx = q.ymx = 0;
  unsigned nv = 0;
  if (p < cfg::NPTS) {
    q = tile_rect(pos2d[2 * p], pos2d[2 * p + 1], radius[p]);
#pragma unroll
    for (int dx = 0; dx < cfg::KWIN; ++dx) {
      const bool vx = (q.xmn + dx) < q.xmx;
#pragma unroll
      for (int dy = 0; dy < cfg::KWIN; ++dy)
        nv += (vx && ((q.ymn + dy) < q.ymx)) ? 1u : 0u;
    }
  }
  s[t] = nv;
  __syncthreads();
  for (int off = 1; off < TPB; off <<= 1) {
    const unsigned u = (t >= off) ? s[t - off] : 0u;
    __syncthreads();
    s[t] += u;
    __syncthreads();
  }
  unsigned pos = g_blkoff[blockIdx.x] + s[t] - nv;
  if (p < cfg::NPTS) {
    const unsigned u = __float_as_uint(depth[p]);
    const unsigned mono = (u >> 31) ? ~u : (u | 0x80000000u);
    const unsigned kd = ~mono;   // ascending kd == descending depth
#pragma unroll
    for (int dx = 0; dx < cfg::KWIN; ++dx) {
      const int xi = q.xmn + dx;
      const bool vx = xi < q.xmx;
#pragma unroll
      for (int dy = 0; dy < cfg::KWIN; ++dy) {
        const int yi = q.ymn + dy;
        if (vx && (yi < q.ymx)) {
          const unsigned tid13 = (unsigned)(xi * cfg::NBH + yi);
          keys[pos++] = ((unsigned long long)tid13 << 51) |
                        ((unsigned long long)kd << 19) |
                        (unsigned long long)(unsigned)p;
        }
      }
    }
  }
}

// ---- K5a: radix histogram; chunk staged by the Tensor Data Mover -----------
__global__ void __launch_bounds__(TPB)
k_radix_hist(const unsigned long long* __restrict__ keys,
             const unsigned* __restrict__ g_M,
             unsigned* __restrict__ g_hist, int shift) {
  __shared__ unsigned long long stage[cfg::CHUNK];   // 32 KB
  __shared__ unsigned h[256];
  const unsigned M = *g_M;
  const int t = threadIdx.x;
  h[t] = 0u;
  const unsigned chunk0 = (unsigned)blockIdx.x * (unsigned)cfg::CHUNK;
  const unsigned rem = (M > chunk0) ? (M - chunk0) : 0u;
  if (t == 0 && rem) {
    tdm_load_1d_b64(&stage[0], keys + chunk0, (unsigned)cfg::CHUNK, rem);
    __builtin_amdgcn_s_wait_tensorcnt(0);            // s_wait_tensorcnt 0
  }
  __syncthreads();
#pragma unroll
  for (int r = 0; r < cfg::IPT; ++r) {
    const unsigned li = (unsigned)(r * TPB + t);
    const unsigned i = chunk0 + li;
    if (i < M) {
      const unsigned d = (unsigned)(stage[li] >> shift) & 255u;
      atomicAdd(&h[d], 1u);
    }
  }
  __syncthreads();
  g_hist[(unsigned)t * (unsigned)cfg::NBLK_R + (unsigned)blockIdx.x] = h[t];
}

// ---- K5b: exclusive scan over the digit-major histogram (in place) ---------
__global__ void __launch_bounds__(1024)
k_radix_scan(unsigned* __restrict__ g_hist) {
  __shared__ unsigned s[1024];
  const int t = threadIdx.x;
  unsigned carry = 0;
  for (unsigned base = 0; base < cfg::HIST_LEN; base += 4096u) {
    unsigned v[4]; unsigned sum = 0;
    const unsigned i0 = base + (unsigned)t * 4u;
#pragma unroll
    for (int j = 0; j < 4; ++j) {
      const unsigned i = i0 + j;
      v[j] = (i < cfg::HIST_LEN) ? g_hist[i] : 0u;
      sum += v[j];
    }
    s[t] = sum;
    __syncthreads();
    for (int off = 1; off < 1024; off <<= 1) {
      const unsigned u = (t >= off) ? s[t - off] : 0u;
      __syncthreads();
      s[t] += u;
      __syncthreads();
    }
    unsigned run = carry + s[t] - sum;
#pragma unroll
    for (int j = 0; j < 4; ++j) {
      const unsigned i = i0 + j;
      if (i < cfg::HIST_LEN) g_hist[i] = run;
      run += v[j];
    }
    const unsigned tot = s[1023];
    __syncthreads();
    carry += tot;
  }
}

// ---- K5c: stable scatter; async global->LDS staging + wave32 ballot rank ---
// Each wave owns a contiguous 512-item sub-chunk so index order == (wave, r, lane).
__global__ void __launch_bounds__(TPB)
k_radix_scatter(const unsigned long long* __restrict__ keysIn,
                unsigned long long* __restrict__ keysOut,
                const unsigned* __restrict__ g_M,
                const unsigned* __restrict__ g_hist, int shift) {
  __shared__ unsigned long long stage[cfg::CHUNK];   // 32 KB
  __shared__ unsigned wcnt[8 * 256];                 // per-wave per-digit counts (8 KB)
  const unsigned M = *g_M;
  const int t = threadIdx.x;
  const int wave = t >> 5;
  const int lane = t & 31;
  const unsigned below = (1u << lane) - 1u;
  for (int j = t; j < 8 * 256; j += TPB) wcnt[j] = 0u;

  const unsigned chunk0 = (unsigned)blockIdx.x * (unsigned)cfg::CHUNK;
  const unsigned wlocal = (unsigned)wave * (32u * (unsigned)cfg::IPT);
  const unsigned long long sbase =
      (unsigned long long)(uintptr_t)keysIn + (unsigned long long)chunk0 * 8ull;

  // Issue all 16 wave-wide async loads up front (no VGPR destinations),
  // then a single ASYNCcnt wait: each wave consumes only its own staged data.
#pragma unroll
  for (int r = 0; r < cfg::IPT; ++r) {
    const unsigned li = wlocal + (unsigned)r * 32u + (unsigned)lane;
    if (chunk0 + li < M) {
      const unsigned lds = (unsigned)(uintptr_t)&stage[li];
      const unsigned off = li * 8u;
      asm volatile("global_load_async_to_lds_b64 %0, %1, %2 offset:0"
                   :
                   : "v"(lds), "v"(off), "s"(sbase)
                   : "memory");
    }
  }
  __syncthreads();                                   // wcnt zeroing visible
  asm volatile("s_wait_asynccnt 0x0" ::: "memory");

  unsigned long long k[cfg::IPT];
  unsigned rk[cfg::IPT];

#pragma unroll
  for (int r = 0; r < cfg::IPT; ++r) {
    const unsigned li = wlocal + (unsigned)r * 32u + (unsigned)lane;
    const bool act = (chunk0 + li) < M;
    k[r] = act ? stage[li] : 0ull;
    const unsigned d = (unsigned)(k[r] >> shift) & 255u;
    unsigned m = 0xFFFFFFFFu;
#pragma unroll
    for (int b = 0; b < 8; ++b) {
      const unsigned bal = ballot32(((d >> b) & 1u) != 0u);
      m &= ((d >> b) & 1u) ? bal : ~bal;
    }
    m &= ballot32(act);
    const unsigned prev = wcnt[wave * 256 + d];        // earlier rounds, this wave
    const unsigned rlocal = __popc(m & below);
    rk[r] = prev + rlocal;
    if (act && rlocal == 0u) atomicAdd(&wcnt[wave * 256 + d], (unsigned)__popc(m));
  }
  __syncthreads();
  // exclusive scan across the 8 waves, per digit (thread t owns digit t)
  {
    unsigned sum = 0;
#pragma unroll
    for (int w = 0; w < 8; ++w) {
      const unsigned v = wcnt[w * 256 + t];
      wcnt[w * 256 + t] = sum;
      sum += v;
    }
  }
  __syncthreads();
#pragma unroll
  for (int r = 0; r < cfg::IPT; ++r) {
    const unsigned li = wlocal + (unsigned)r * 32u + (unsigned)lane;
    if (chunk0 + li < M) {
      const unsigned d = (unsigned)(k[r] >> shift) & 255u;
      const unsigned dst = g_hist[d * (unsigned)cfg::NBLK_R + (unsigned)blockIdx.x] +
                           wcnt[wave * 256 + d] + rk[r];
      keysOut[dst] = k[r];
    }
  }
}

// ---- K6: emit tile_indices / sorted_depth (tail padded -1 / 0.0) -----------
__global__ void __launch_bounds__(TPB)
k_write_out(const unsigned long long* __restrict__ keys,
            const float* __restrict__ depth,
            const unsigned* __restrict__ g_M,
            int* __restrict__ tile_indices,
            float* __restrict__ sorted_depth) {
  const unsigned i = (unsigned)blockIdx.x * TPB + threadIdx.x;
  if (i >= cfg::EI) return;
  const unsigned M = *g_M;
  if (i < M) {
    const unsigned long long key = keys[i];
    const unsigned pid = (unsigned)(key & 0x7FFFFull);
    tile_indices[i] = (int)pid;
    sorted_depth[i] = depth[pid];
  } else {
    tile_indices[i] = -1;
    sorted_depth[i] = 0.0f;
  }
}

extern "C" void kernel_launch(void* const* d_in, const int* in_sizes, int n_in,
                              void* d_out, int out_size, void* d_ws, size_t ws_size,
                              hipStream_t stream) {
  (void)in_sizes; (void)n_in; (void)out_size; (void)ws_size;
  const float* pos2d  = (const float*)d_in[0];
  const float* radius = (const float*)d_in[1];
  const float* depth  = (const float*)d_in[2];

  char* ws = (char*)d_ws;
  size_t off = 0;
  auto take = [&](size_t bytes) -> void* {
    void* p = ws + off;
    off = (off + bytes + 255) & ~(size_t)255;
    return p;
  };
  unsigned long long* keysA = (unsigned long long*)take((size_t)cfg::EI * 8);
  unsigned long long* keysB = (unsigned long long*)take((size_t)cfg::EI * 8);
  unsigned* g_hist   = (unsigned*)take((size_t)cfg::HIST_LEN * 4);
  unsigned* g_counts = (unsigned*)take((size_t)cfg::NB * 4);
  unsigned* g_blk    = (unsigned*)take((size_t)cfg::NBLK_A * 4);
  unsigned* g_M      = (unsigned*)take(4);

  int*   tile_count   = (int*)d_out;
  int*   tile_indices = tile_count + cfg::NB;
  float* sorted_depth = (float*)(tile_indices + cfg::EI);

  k_zero<<<dim3((cfg::NB + TPB - 1) / TPB), dim3(TPB), 0, stream>>>(g_counts);
  k_tile_hist<<<dim3(cfg::NBLK_A), dim3(TPB), 0, stream>>>(pos2d, radius, g_counts, g_blk);
  k_scan_blocks<<<dim3(1), dim3(TPB), 0, stream>>>(g_blk, g_M);
  k_scan_tiles<<<dim3(1), dim3(TPB), 0, stream>>>(g_counts, tile_count);
  k_emit<<<dim3(cfg::NBLK_A), dim3(TPB), 0, stream>>>(pos2d, radius, depth, g_blk, keysA);

  unsigned long long* bufIn  = keysA;
  unsigned long long* bufOut = keysB;
  for (int p = 0; p < cfg::PASSES; ++p) {
    const int shift = cfg::SHIFT0 + 8 * p;
    k_radix_hist<<<dim3(cfg::NBLK_R), dim3(TPB), 0, stream>>>(bufIn, g_M, g_hist, shift);
    k_radix_scan<<<dim3(1), dim3(1024), 0, stream>>>(g_hist);
    k_radix_scatter<<<dim3(cfg::NBLK_R), dim3(TPB), 0, stream>>>(bufIn, bufOut, g_M, g_hist, shift);
    unsigned long long* tmp = bufIn; bufIn = bufOut; bufOut = tmp;
  }
  // 6 passes -> result back in keysA (bufIn)
  k_write_out<<<dim3((cfg::EI + TPB - 1) / TPB), dim3(TPB), 0, stream>>>(
      bufIn, depth, g_M, tile_indices, sorted_depth);
}